// MultilayerRNN_23742579213120
// MI455X (gfx1250) — compile-verified
//
#include <hip/hip_runtime.h>
#include <hip/hip_bf16.h>

typedef __attribute__((ext_vector_type(2))) float v2f;
typedef __attribute__((ext_vector_type(8))) float v8f;

#define DI 512
#define DH 1024
#define DO 128
#define BB 64
#define SS 512
#define LDW1 (DI + DH)   // 1536
#define LDW2 (DH + DH)   // 2048
#define MROWS (BB * SS)  // 32768

// D = A(16x4,f32) * B(4x16,f32) + C(16x16,f32), full fp32 — faithful to reference.
__device__ __forceinline__ v8f wmma4(v2f a, v2f b, v8f c) {
  return __builtin_amdgcn_wmma_f32_16x16x4_f32(
      /*neg_a=*/false, a, /*neg_b=*/false, b,
      /*c_mod=*/(short)0, c, /*reuse_a=*/false, /*reuse_b=*/false);
}

// Async global->LDS 128-bit copy (CDNA5 GLOBAL_LOAD_ASYNC_TO_LDS_B128,
// tracked by ASYNCcnt). Per-lane: copies 16 bytes from global gptr to the LDS address in lds_ptr.
__device__ __forceinline__ void async_copy_b128(void* lds_ptr, const void* gptr) {
  unsigned lds_off = (unsigned)(uintptr_t)lds_ptr;   // low 32 bits of the pointer are the LDS offset
  asm volatile("global_load_async_to_lds_b128 %0, %1, off"
               :: "v"(lds_off), "v"(gptr) : "memory");
}
__device__ __forceinline__ void async_wait0() {
  asm volatile("s_wait_asynccnt 0x0" ::: "memory");
}

// Load contiguous K-pair fragments (lanes 0-15: K={k0,k0+1}; 16-31: K={k0+2,k0+3})
__device__ __forceinline__ v2f frag2(const float* p) { return *(const v2f*)p; }

// ---------------------------------------------------------------------------
// Kernel 1: fused embedding gather + xp1 = x @ Wx1 + b1
//   one workgroup (8 waves) per 16-row M-tile; each wave covers 8 N-tiles.
// ---------------------------------------------------------------------------
__global__ __launch_bounds__(256)
void k_embed_gemm(const int* __restrict__ tokens,
                  const float* __restrict__ emb,
                  const float* __restrict__ W1,
                  const float* __restrict__ b1,
                  float* __restrict__ xp)    // [MROWS, DH]
{
  __shared__ float As[16 * (DI + 4)];        // row pitch 516 floats = 2064 B (16B mult)
  __shared__ int   tok[16];
  const int tid  = threadIdx.x;
  const int lane = tid & 31;
  const int wave = tid >> 5;
  const int m0   = blockIdx.x * 16;

  if (tid < 16) tok[tid] = tokens[m0 + tid];
  __syncthreads();
  // gather 16 embedding rows straight into LDS (async, no VGPR round-trip)
  for (int idx = tid * 4; idx < 16 * DI; idx += 256 * 4) {
    int r = idx >> 9;            // /DI
    int c = idx & (DI - 1);
    async_copy_b128(&As[r * (DI + 4) + c], &emb[(long)tok[r] * DI + c]);
  }
  async_wait0();
  __syncthreads();

  const int mrow  = lane & 15;
  const int khalf = lane >> 4;       // 0 or 1
  for (int j = 0; j < 8; ++j) {
    const int nt = wave * 8 + j;
    const int n  = nt * 16 + (lane & 15);
    const float* wn = W1 + (long)n * LDW1;   // row n of W1; Wx part = cols [0,DI)
    v8f acc = {};
    for (int k0 = 0; k0 < DI; k0 += 4) {
      v2f a = frag2(&As[mrow * (DI + 4) + k0 + khalf * 2]);
      v2f b = frag2(wn + k0 + khalf * 2);
      acc = wmma4(a, b, acc);
    }
    const float bias = b1[n];
    for (int r = 0; r < 8; ++r) {
      int m = m0 + r + khalf * 8;
      xp[(long)m * DH + n] = acc[r] + bias;
    }
  }
}

// ---------------------------------------------------------------------------
// Kernel 2: xp2 = h1 @ Wx2 + b2   (A tile staged to LDS via async copies)
// ---------------------------------------------------------------------------
__global__ __launch_bounds__(256)
void k_gemm_bias(const float* __restrict__ Ain,   // [MROWS, DH]
                 const float* __restrict__ W,     // [DH, LDW2], Wx = cols [0,DH)
                 const float* __restrict__ bias,  // [DH]
                 float* __restrict__ out)         // [MROWS, DH]
{
  extern __shared__ float As[];                   // 16 x (DH+4); pitch 4112 B
  const int tid  = threadIdx.x;
  const int lane = tid & 31;
  const int wave = tid >> 5;
  const int m0   = blockIdx.x * 16;
  const int AP   = DH + 4;

  for (int idx = tid * 4; idx < 16 * DH; idx += 256 * 4) {
    int r = idx >> 10;
    int c = idx & (DH - 1);
    async_copy_b128(&As[r * AP + c], &Ain[(long)(m0 + r) * DH + c]);
  }
  async_wait0();
  __syncthreads();

  const int mrow  = lane & 15;
  const int khalf = lane >> 4;
  for (int j = 0; j < 8; ++j) {
    const int nt = wave * 8 + j;
    const int n  = nt * 16 + (lane & 15);
    const float* wn = W + (long)n * LDW2;
    v8f acc = {};
    for (int k0 = 0; k0 < DH; k0 += 4) {
      v2f a = frag2(&As[mrow * AP + k0 + khalf * 2]);
      v2f b = frag2(wn + k0 + khalf * 2);
      acc = wmma4(a, b, acc);
    }
    const float bv = bias[n];
    for (int r = 0; r < 8; ++r) {
      int m = m0 + r + khalf * 8;
      out[(long)m * DH + n] = acc[r] + bv;
    }
  }
}

// ---------------------------------------------------------------------------
// Kernel 3: linear recurrence, in place:  h_t = xp_t + h_{t-1} @ Wh
//   One workgroup (32 waves) per 16 batch rows; H (16xDH f32) lives in LDS
//   for all 512 steps; Wh streamed from L2 each step. h_0 = xp_0 (ref scan).
// ---------------------------------------------------------------------------
__global__ __launch_bounds__(1024)
void k_rnn(float* __restrict__ xp,       // [BB, SS, DH]; in: xp, out: h
           const float* __restrict__ W,  // Wh[k][n] = W[n*ldw + koff + k]
           int ldw, int koff)
{
  extern __shared__ float Hs[];          // 16 x (DH+4); pitch 4112 B
  const int tid  = threadIdx.x;
  const int lane = tid & 31;
  const int wave = tid >> 5;             // 0..31
  const int b0   = blockIdx.x * 16;
  const int HP   = DH + 4;

  // h_0 = xp[:, 0, :] (already correct in global memory)
  for (int idx = tid * 4; idx < 16 * DH; idx += 1024 * 4) {
    int r = idx >> 10;
    int c = idx & (DH - 1);
    async_copy_b128(&Hs[r * HP + c], &xp[((long)(b0 + r) * SS) * DH + c]);
  }
  async_wait0();
  __syncthreads();

  const int mrow  = lane & 15;
  const int khalf = lane >> 4;
  const int n0    = wave * 32 + (lane & 15);
  const int n1    = n0 + 16;
  const float* w0 = W + (long)n0 * ldw + koff;
  const float* w1 = W + (long)n1 * ldw + koff;

  for (int t = 1; t < SS; ++t) {
    v8f acc0 = {}, acc1 = {};
    for (int k0 = 0; k0 < DH; k0 += 4) {
      v2f a   = frag2(&Hs[mrow * HP + k0 + khalf * 2]);
      v2f bf0 = frag2(w0 + k0 + khalf * 2);
      v2f bf1 = frag2(w1 + k0 + khalf * 2);
      acc0 = wmma4(a, bf0, acc0);
      acc1 = wmma4(a, bf1, acc1);
    }
    __syncthreads();                     // all reads of Hs complete
    for (int r = 0; r < 8; ++r) {
      int m = r + khalf * 8;
      long g = ((long)(b0 + m) * SS + t) * DH;
      float h0 = xp[g + n0] + acc0[r];
      float h1 = xp[g + n1] + acc1[r];
      Hs[m * HP + n0] = h0;  xp[g + n0] = h0;
      Hs[m * HP + n1] = h1;  xp[g + n1] = h1;
    }
    __syncthreads();
  }
}

// ---------------------------------------------------------------------------
// Kernel 4: fc + log_softmax: out = log_softmax(h2 @ fcW.T + fcb)
//   8 waves, each owns one 16-col N-tile of the 128 outputs.
// ---------------------------------------------------------------------------
__global__ __launch_bounds__(256)
void k_fc_logsoftmax(const float* __restrict__ H,    // [MROWS, DH]
                     const float* __restrict__ fcW,  // [DO, DH]
                     const float* __restrict__ fcb,  // [DO]
                     float* __restrict__ out)        // [MROWS, DO]
{
  extern __shared__ float As[];                      // 16 x (DH+4)
  __shared__ float Ls[16 * (DO + 4)];
  const int tid  = threadIdx.x;
  const int lane = tid & 31;
  const int wave = tid >> 5;
  const int m0   = blockIdx.x * 16;
  const int AP   = DH + 4;
  const int LP   = DO + 4;

  for (int idx = tid * 4; idx < 16 * DH; idx += 256 * 4) {
    int r = idx >> 10;
    int c = idx & (DH - 1);
    async_copy_b128(&As[r * AP + c], &H[(long)(m0 + r) * DH + c]);
  }
  async_wait0();
  __syncthreads();

  const int mrow  = lane & 15;
  const int khalf = lane >> 4;
  const int n     = wave * 16 + (lane & 15);
  const float* wn = fcW + (long)n * DH;
  v8f acc = {};
  for (int k0 = 0; k0 < DH; k0 += 4) {
    v2f a = frag2(&As[mrow * AP + k0 + khalf * 2]);
    v2f b = frag2(wn + k0 + khalf * 2);
    acc = wmma4(a, b, acc);
  }
  const float bv = fcb[n];
  for (int r = 0; r < 8; ++r)
    Ls[(r + khalf * 8) * LP + n] = acc[r] + bv;
  __syncthreads();

  if (tid < 16) {
    const int r = tid;
    float mx = Ls[r * LP];
    for (int o = 1; o < DO; ++o) mx = fmaxf(mx, Ls[r * LP + o]);
    float s = 0.0f;
    for (int o = 0; o < DO; ++o) s += expf(Ls[r * LP + o] - mx);
    const float lse = mx + logf(s);
    for (int o = 0; o < DO; ++o)
      out[(long)(m0 + r) * DO + o] = Ls[r * LP + o] - lse;
  }
}

// ---------------------------------------------------------------------------
extern "C" void kernel_launch(void* const* d_in, const int* in_sizes, int n_in,
                              void* d_out, int out_size, void* d_ws, size_t ws_size,
                              hipStream_t stream)
{
  const int*   tokens = (const int*)  d_in[0];
  const float* emb    = (const float*)d_in[1];
  const float* W1     = (const float*)d_in[2];
  const float* b1     = (const float*)d_in[3];
  const float* W2     = (const float*)d_in[4];
  const float* b2     = (const float*)d_in[5];
  const float* fcW    = (const float*)d_in[6];
  const float* fcb    = (const float*)d_in[7];
  float* out = (float*)d_out;

  float* xp1 = (float*)d_ws;                     // [MROWS, DH] fp32 (xp1 then h1)
  float* xp2 = xp1 + (size_t)MROWS * DH;         // [MROWS, DH] fp32 (xp2 then h2)

  const size_t lds_a = (size_t)16 * (DH + 4) * sizeof(float);

  k_embed_gemm   <<<MROWS / 16, 256, 0,     stream>>>(tokens, emb, W1, b1, xp1);
  k_rnn          <<<BB / 16,   1024, lds_a, stream>>>(xp1, W1, LDW1, DI);
  k_gemm_bias    <<<MROWS / 16, 256, lds_a, stream>>>(xp1, W2, b2, xp2);
  k_rnn          <<<BB / 16,   1024, lds_a, stream>>>(xp2, W2, LDW2, DH);
  k_fc_logsoftmax<<<MROWS / 16, 256, lds_a, stream>>>(xp2, fcW, fcb, out);
}